// Expert_31301721653487
// MI455X (gfx1250) — compile-verified
//
#include <hip/hip_runtime.h>

// Problem constants (from reference): B=32, L=4096, H=1024, N=4
#define BATCH   32
#define SEQ     4096
#define HID     1024
#define KDIM    4096   // N*H
#define NOUT    4
#define KSPLIT  8      // K-dim split across blockIdx.y: 512 blocks -> full chip
#define KSLICE  (KDIM / KSPLIT)          // 512 per block
#define KWAVE   (KSLICE / 8)             // 64 per wave
#define KGROUP  32                       // k per load/compute phase (8 WMMA pairs)

typedef __attribute__((ext_vector_type(2))) float v2f;
typedef __attribute__((ext_vector_type(4))) float v4f;
typedef __attribute__((ext_vector_type(8))) float v8f;

// GEMM1: Y(32x1024) = X(32x4096) @ W1^T(4096x1024) + b1
//   X[b,k] = hidden[b*L*H + pos[b]*H + k]   (gather fused into base offset)
// Grid: (64 N-tiles) x (8 K-slices), 256 threads = 8 waves per block.
// Each wave covers KWAVE=64 of K and accumulates BOTH batch M-tiles
// (rows 0-15 and 16-31) so each W1 element is read exactly once chip-wide;
// 512 blocks spread the 16.8MB W1 stream across the whole memory system.
// Inner loop is phase-split: burst 24 global_load_b64 (one clause, deep
// loadcnt), then 16 back-to-back V_WMMA_F32_16X16X4_F32.
__global__ __launch_bounds__(256)
void gemm1_wmma(const float* __restrict__ hidden,
                const int*   __restrict__ pos,
                const float* __restrict__ W1,
                const float* __restrict__ b1,
                float*       __restrict__ Ypart)   // (KSPLIT, 32, 1024)
{
    const int n_tile = blockIdx.x;       // 0..63
    const int kslice = blockIdx.y;       // 0..KSPLIT-1
    const int tid    = threadIdx.x;      // 0..255
    const int wave   = tid >> 5;         // 0..7
    const int lane   = tid & 31;
    const int lane_lo = lane & 15;       // M (for A) / N (for B) within tile
    const int lane_hi = lane >> 4;       // selects K pair {0,1} vs {2,3}

    // Loop-invariant fragment base pointers, already offset by this wave's
    // K-slab and the lane's K-pair: unrolled body addresses purely via
    // immediate offsets.
    const int kbase = kslice * KSLICE + wave * KWAVE + 2 * lane_hi;
    const int m0 = lane_lo;
    const int m1 = 16 + lane_lo;
    const float* aPtr0 = hidden + ((long)m0 * SEQ + (long)pos[m0]) * HID + kbase;
    const float* aPtr1 = hidden + ((long)m1 * SEQ + (long)pos[m1]) * HID + kbase;
    const float* bPtr  = W1 + (long)(n_tile * 16 + lane_lo) * KDIM + kbase;

    v8f acc0 = {};
    v8f acc1 = {};

    // Per ISA layout (32-bit A 16x4 / B 4x16): VGPR0 holds K=k0 (low lanes)
    // or K=k0+2 (high lanes); VGPR1 holds the next K -> contiguous float2.
    for (int kk = 0; kk < KWAVE; kk += KGROUP) {
        v2f a0v[KGROUP / 4], a1v[KGROUP / 4], bbv[KGROUP / 4];
        // Phase 1: burst all loads (registers; fully unrolled const indices)
        #pragma unroll
        for (int u = 0; u < KGROUP / 4; ++u) {
            const int off = kk + 4 * u;
            a0v[u] = *reinterpret_cast<const v2f*>(aPtr0 + off);
            a1v[u] = *reinterpret_cast<const v2f*>(aPtr1 + off);
            bbv[u] = *reinterpret_cast<const v2f*>(bPtr  + off);
        }
        // Phase 2: back-to-back WMMAs; loadcnt waits count down across them
        #pragma unroll
        for (int u = 0; u < KGROUP / 4; ++u) {
            acc0 = __builtin_amdgcn_wmma_f32_16x16x4_f32(
                       false, a0v[u], false, bbv[u], (short)0, acc0, false, false);
            acc1 = __builtin_amdgcn_wmma_f32_16x16x4_f32(
                       false, a1v[u], false, bbv[u], (short)0, acc1, false, false);
        }
    }

    // Cross-wave K reduction through LDS: 2 tiles * 8 waves * 256 floats = 16KB
    __shared__ float red[2][8][32][8];
    *reinterpret_cast<v8f*>(&red[0][wave][lane][0]) = acc0;
    *reinterpret_cast<v8f*>(&red[1][wave][lane][0]) = acc1;
    __syncthreads();

    // 256 threads finalize 512 outputs (2 each). D-tile layout:
    // VGPR r, lane l -> M = r + 8*(l>=16), N = l%16.
    const int r     = tid >> 5;       // accumulator VGPR index 0..7
    const int rlane = tid & 31;
    const int m_loc = r + 8 * (rlane >> 4);
    const int n_loc = rlane & 15;
    const int gn    = n_tile * 16 + n_loc;
    const float bias = (kslice == 0) ? b1[gn] : 0.0f;   // bias added once
    float* Yslice = Ypart + (long)kslice * (BATCH * HID);
    #pragma unroll
    for (int mt = 0; mt < 2; ++mt) {
        float sum = bias;
        #pragma unroll
        for (int w = 0; w < 8; ++w)
            sum += red[mt][w][rlane][r];
        Yslice[(mt * 16 + m_loc) * HID + gn] = sum;
    }
}

// GEMM2: out(32x4) = (sum of KSPLIT Y partials)(32x1024) @ W2^T(1024x4) + b2.
// Tiny; one block, one thread per output element, float4 dot products.
__global__ __launch_bounds__(128)
void gemm2_dot(const float* __restrict__ Ypart,   // (KSPLIT, 32, 1024)
               const float* __restrict__ W2,
               const float* __restrict__ b2,
               float*       __restrict__ out)
{
    const int t = threadIdx.x;   // 0..127
    const int b = t >> 2;
    const int n = t & 3;
    const float* y = Ypart + b * HID;
    const float* w = W2 + n * HID;
    float s = 0.f;
    for (int h = 0; h < HID; h += 4) {
        v4f yv = {};
        #pragma unroll
        for (int ks = 0; ks < KSPLIT; ++ks)
            yv += *reinterpret_cast<const v4f*>(y + (long)ks * (BATCH * HID) + h);
        v4f wv = *reinterpret_cast<const v4f*>(w + h);
        s += yv.x * wv.x + yv.y * wv.y + yv.z * wv.z + yv.w * wv.w;
    }
    out[b * NOUT + n] = s + b2[n];
}

extern "C" void kernel_launch(void* const* d_in, const int* in_sizes, int n_in,
                              void* d_out, int out_size, void* d_ws, size_t ws_size,
                              hipStream_t stream) {
    const float* hidden = (const float*)d_in[0];   // (32, 4096, 1024) f32
    const int*   pos    = (const int*)  d_in[1];   // (32, 1) i32
    const float* W1     = (const float*)d_in[2];   // (1024, 4096) f32
    const float* b1     = (const float*)d_in[3];   // (1024,) f32
    const float* W2     = (const float*)d_in[4];   // (4, 1024) f32
    const float* b2     = (const float*)d_in[5];   // (4,) f32
    float* out   = (float*)d_out;                  // (32, 4) f32
    float* Ypart = (float*)d_ws;                   // (KSPLIT, 32, 1024) f32 = 1MB scratch

    gemm1_wmma<<<dim3(64, KSPLIT), dim3(256), 0, stream>>>(hidden, pos, W1, b1, Ypart);
    gemm2_dot<<<dim3(1), dim3(128), 0, stream>>>(Ypart, W2, b2, out);
}